// GraphAggregation_65481071404067
// MI455X (gfx1250) — compile-verified
//
#include <hip/hip_runtime.h>

typedef float v2f __attribute__((ext_vector_type(2)));
typedef float v8f __attribute__((ext_vector_type(8)));

#define NODES 196
#define VDIM 4
#define ROWS_PER_BLOCK 128      // 8 waves x 16 rows
#define CHUNK 28                // nodes staged per LDS chunk (196 = 7*28)
#define NCHUNK 7
#define LDS_ROW_F 116           // floats per LDS row: 28*4 data + 4 pad (bank stagger)

// out[b][d] = (1/196) * sum_n X[b][n][d], computed as a skinny GEMM on
// v_wmma_f32_16x16x4_f32:  D(16x16) += A(16x4 = 16 batch rows x one node's
// 4 components) * B(4x16 = (1/196)*I4 in the top-left), accumulated over all
// 196 nodes. Only columns 0..3 of D are used.
__global__ __launch_bounds__(256)
void GraphAggregation_mean196x4_wmma(const float* __restrict__ X,
                                     float* __restrict__ out, int B) {
  __shared__ __align__(16) float lds[ROWS_PER_BLOCK * LDS_ROW_F]; // 59392 B

  const int t    = threadIdx.x;
  const int lane = t & 31;
  const int w    = t >> 5;        // wave id 0..7
  const int m    = lane & 15;     // row-in-tile / N-column index
  const int half = lane >> 4;     // 0: K={0,1}, 1: K={2,3}
  const int b0   = blockIdx.x * ROWS_PER_BLOCK;

  // B fragment = (1/196) * I4.  32-bit B(4x16) layout (mirrors C/D striping):
  // VGPR0 holds row K=2*half+0, VGPR1 holds row K=2*half+1, N = lane%16.
  const float s = 1.0f / 196.0f;
  v2f bfrag;
  bfrag.x = (m == 2 * half + 0) ? s : 0.0f;
  bfrag.y = (m == 2 * half + 1) ? s : 0.0f;

  v8f acc = {};

  const float4* __restrict__ Xv = (const float4*)X; // one float4 == one node vector

  for (int c = 0; c < NCHUNK; ++c) {
    // ---- stage 128 rows x 28 node-vectors into LDS, coalesced b128 loads ----
#pragma unroll
    for (int i = 0; i < (ROWS_PER_BLOCK * CHUNK) / 256; ++i) { // 14 iters
      int idx = t + i * 256;          // 0..3583
      int row = idx / CHUNK;
      int j   = idx - row * CHUNK;
      int gr  = b0 + row;
      if (gr >= B) gr = B - 1;        // benign clamp (B % 128 == 0 in practice)
      float4 v = Xv[(size_t)gr * NODES + c * CHUNK + j];
      *(float4*)&lds[row * LDS_ROW_F + j * 4] = v;
    }
    __syncthreads();

    // ---- 28 WMMAs: A(16x4) per node from LDS (conflict-free ds_load_b64) ----
    const float* arow = &lds[(w * 16 + m) * LDS_ROW_F + half * 2];
#pragma unroll
    for (int n = 0; n < CHUNK; ++n) {
      v2f a = *(const v2f*)(arow + n * 4);
      acc = __builtin_amdgcn_wmma_f32_16x16x4_f32(
          /*neg_a=*/false, a, /*neg_b=*/false, bfrag,
          /*c_mod=*/(short)0, acc, /*reuse_a=*/false, /*reuse_b=*/false);
    }
    __syncthreads();
  }

  // ---- D element (mrow, d): VGPR mrow%8, lane d + 16*(mrow/8).  Need d<4. ----
  if (m < VDIM) {
    int baseRow = b0 + w * 16 + half * 8;
#pragma unroll
    for (int r = 0; r < 8; ++r) {
      int gr = baseRow + r;
      if (gr < B) out[(size_t)gr * VDIM + m] = acc[r];
    }
  }
}

extern "C" void kernel_launch(void* const* d_in, const int* in_sizes, int n_in,
                              void* d_out, int out_size, void* d_ws, size_t ws_size,
                              hipStream_t stream) {
  (void)n_in; (void)d_ws; (void)ws_size; (void)out_size;
  const float* X = (const float*)d_in[0];
  float* out = (float*)d_out;
  const int B = in_sizes[0] / (NODES * VDIM);           // 131072
  const int nblocks = (B + ROWS_PER_BLOCK - 1) / ROWS_PER_BLOCK; // 1024
  GraphAggregation_mean196x4_wmma<<<nblocks, 256, 0, stream>>>(X, out, B);
}